// QuantumLayer_64888365908273
// MI455X (gfx1250) — compile-verified
//
#include <hip/hip_runtime.h>
#include <math.h>

// ---------------------------------------------------------------------------
// out = Z-expectations of  U · (⊗_q RY(x_q)|0>),  U batch-independent.
// Kernel 1: build ReU^T / ImU^T as separate planes  plane[n][k]  (512 KB ws).
// Kernel 2: B×256 real GEMM on v_wmma_f32_16x16x4_f32; A fragments generated
//           on the fly from the 4-qubit tensor-product factors; Z-expectation
//           reduction via a 16-lane Walsh–Hadamard butterfly (ds_swizzle).
// ---------------------------------------------------------------------------

typedef float v2f __attribute__((ext_vector_type(2)));
typedef float v8f __attribute__((ext_vector_type(8)));

#define NQ   8
#define DIM  256
#define MBLK 4              // M-tiles (of 16 batch rows) per block
#define ROWS (MBLK * 16)    // 64 batch rows per block

// ---------------- Kernel 1: build the circuit unitary ----------------------

__device__ __forceinline__ void ry_gate(float2* st, int w, float t, int tid) {
    int stride = 1 << (7 - w);
    if (tid < 128) {
        int low = tid & (stride - 1);
        int i0  = ((tid - low) << 1) | low;
        int i1  = i0 + stride;
        float s, c;
        sincosf(0.5f * t, &s, &c);
        float2 a0 = st[i0], a1 = st[i1];
        st[i0] = make_float2(c * a0.x - s * a1.x, c * a0.y - s * a1.y);
        st[i1] = make_float2(s * a0.x + c * a1.x, s * a0.y + c * a1.y);
    }
    __syncthreads();
}

__device__ __forceinline__ void rz_gate(float2* st, int w, float t, int tid) {
    int stride = 1 << (7 - w);
    if (tid < 128) {
        int low = tid & (stride - 1);
        int i0  = ((tid - low) << 1) | low;
        int i1  = i0 + stride;
        float s, c;
        sincosf(0.5f * t, &s, &c);
        float2 a0 = st[i0], a1 = st[i1];
        st[i0] = make_float2(a0.x * c + a0.y * s, a0.y * c - a0.x * s); // * e^{-it/2}
        st[i1] = make_float2(a1.x * c - a1.y * s, a1.y * c + a1.x * s); // * e^{+it/2}
    }
    __syncthreads();
}

__device__ __forceinline__ void cnot_gate(float2* st, int c, int t, int tid) {
    int stride = 1 << (7 - t);
    if (tid < 128) {
        int low = tid & (stride - 1);
        int i0  = ((tid - low) << 1) | low;
        int i1  = i0 + stride;
        if ((i0 >> (7 - c)) & 1) {
            float2 tmp = st[i0]; st[i0] = st[i1]; st[i1] = tmp;
        }
    }
    __syncthreads();
}

__global__ __launch_bounds__(256)
void build_unitary_kernel(const float* __restrict__ conv,
                          const float* __restrict__ pool,
                          float* __restrict__ RePlane,   // [n][k] = Re U[n][k]
                          float* __restrict__ ImPlane) { // [n][k] = Im U[n][k]
    __shared__ float2 st[DIM];
    const int tid = threadIdx.x;
    const int j   = blockIdx.x;          // basis column k = j
    st[tid] = make_float2(tid == j ? 1.0f : 0.0f, 0.0f);
    __syncthreads();

    for (int l = 0; l < 2; ++l) {
        for (int i = 0; i < 7; ++i) {
            const float* g = conv + (l * 7 + i) * 3;
            rz_gate(st, i,     g[0], tid);
            ry_gate(st, i + 1, g[1], tid);
            cnot_gate(st, i, i + 1, tid);
            ry_gate(st, i + 1, g[2], tid);
        }
    }
    const int srcs[7] = {0, 1, 2, 3, 0, 1, 0};
    const int snks[7] = {4, 5, 6, 7, 2, 3, 1};
    const int pidx[7] = {0, 0, 0, 0, 1, 1, 2};
    for (int g = 0; g < 7; ++g) {
        const float* pp = pool + pidx[g] * 3;
        rz_gate(st, srcs[g], pp[0], tid);
        ry_gate(st, snks[g], pp[1], tid);
        cnot_gate(st, srcs[g], snks[g], tid);
        ry_gate(st, snks[g], pp[2], tid);
    }
    // state[n] = U[n][j]  ->  plane[n][j]
    RePlane[tid * DIM + j] = st[tid].x;
    ImPlane[tid * DIM + j] = st[tid].y;
}

// ---------------- Kernel 2: WMMA GEMM + Walsh-Hadamard Z-reduce -------------

#define WHT_STAGE(p, D, IMM)                                                   \
    {                                                                          \
        float o_ = __int_as_float(                                             \
            __builtin_amdgcn_ds_swizzle(__float_as_int(p), (IMM)));            \
        p = (lane & (D)) ? (o_ - p) : (p + o_);                                \
    }

__global__ __launch_bounds__(512)
void qcnn_wmma_kernel(const float* __restrict__ x,
                      const float* __restrict__ RePlane,
                      const float* __restrict__ ImPlane,
                      float* __restrict__ out) {
    __shared__ float csC[ROWS][NQ];          // 2 KB
    __shared__ float csS[ROWS][NQ];          // 2 KB
    __shared__ float hiP[ROWS][16];          // 4 KB  product over wires 0-3
    __shared__ float loP[ROWS][16];          // 4 KB  product over wires 4-7
    __shared__ float partial[16][ROWS][5];   // 20 KB WHT coeffs per wave/row

    const int tid = threadIdx.x;
    const int rb  = blockIdx.x * ROWS;       // first batch row of this block

    // --- phase 1: half-angle cos/sin (64 rows x 8 qubits = 512 values) ---
    {
        int row = tid >> 3, q = tid & 7;
        float s, c;
        sincosf(0.5f * x[(rb + row) * NQ + q], &s, &c);
        csC[row][q] = c;
        csS[row][q] = s;
    }
    __syncthreads();

    // --- phase 2: 4-qubit tensor-product factors (2 x 64 x 16 values) ---
    for (int idx = tid; idx < 2 * ROWS * 16; idx += 512) {
        int half = idx >> 10;                // 0: wires 0-3, 1: wires 4-7
        int row  = (idx >> 4) & (ROWS - 1);
        int jj   = idx & 15;
        float p = 1.0f;
#pragma unroll
        for (int b = 0; b < 4; ++b) {
            int q   = half * 4 + b;
            int bit = (jj >> (3 - b)) & 1;
            p *= bit ? csS[row][q] : csC[row][q];
        }
        (half ? loP : hiP)[row][jj] = p;
    }
    __syncthreads();

    // --- GEMM: state(64x256, re+im) = a(64x256) * U^T(256x256) ---
    const int lane = tid & 31;
    const int wv   = tid >> 5;                 // 0..15 -> N-tile
    const int m    = lane & 15;                // A-row within an M-tile
    const int koff = (lane >> 4) << 1;         // K sub-offset: 0 or 2
    const int ncol = (wv << 4) + (lane & 15);  // global N column
    const float* Rb = RePlane + (size_t)ncol * DIM;  // contiguous over k
    const float* Ib = ImPlane + (size_t)ncol * DIM;

    // preload the 4 lo-pairs this lane ever needs, per M-tile
    v2f lo2[MBLK][4];
#pragma unroll
    for (int mt = 0; mt < MBLK; ++mt)
#pragma unroll
        for (int t = 0; t < 4; ++t)
            lo2[mt][t] = *(const v2f*)&loP[mt * 16 + m][koff + 4 * t];

    v8f accR[MBLK], accI[MBLK];
#pragma unroll
    for (int mt = 0; mt < MBLK; ++mt) {
        accR[mt] = (v8f){0.f,0.f,0.f,0.f,0.f,0.f,0.f,0.f};
        accI[mt] = (v8f){0.f,0.f,0.f,0.f,0.f,0.f,0.f,0.f};
    }

#pragma unroll 4
    for (int kb = 0; kb < DIM; kb += 4) {
        const int k    = kb + koff;
        const int hidx = kb >> 4;              // hi index (uniform per step)
        const int t    = (kb >> 2) & 3;        // lo-pair selector
        v2f br = *(const v2f*)&Rb[k];          // {ReU^T[k][n], ReU^T[k+1][n]}
        v2f bi = *(const v2f*)&Ib[k];
#pragma unroll
        for (int mt = 0; mt < MBLK; ++mt) {
            float h = hiP[mt * 16 + m][hidx];
            v2f a;
            a.x = lo2[mt][t].x * h;            // A[row][k]
            a.y = lo2[mt][t].y * h;            // A[row][k+1]
            accR[mt] = __builtin_amdgcn_wmma_f32_16x16x4_f32(
                false, a, false, br, (short)0, accR[mt], false, false);
            accI[mt] = __builtin_amdgcn_wmma_f32_16x16x4_f32(
                false, a, false, bi, (short)0, accI[mt], false, false);
        }
    }

    // --- per-wave Z partial sums via 16-lane Walsh-Hadamard butterfly ---
    // n = wv*16 + (lane&15): qubits 0-3 sign is wave-uniform (bits of wv),
    // qubits 4-7 sign depends on the low 4 bits of n -> WHT across lanes.
    // After WHT, lane pattern l (l = lane&15) holds sum_n (-1)^popc(l&n) p(n):
    //   l=0 -> plain sum (for qubits 0-3), l=8,4,2,1 -> qubits 4,5,6,7.
#pragma unroll
    for (int mt = 0; mt < MBLK; ++mt) {
#pragma unroll
        for (int r = 0; r < 8; ++r) {
            float p = accR[mt][r] * accR[mt][r] + accI[mt][r] * accI[mt][r];
            WHT_STAGE(p, 1, 0x041F)            // ds_swizzle xor 1
            WHT_STAGE(p, 2, 0x081F)            // xor 2
            WHT_STAGE(p, 4, 0x101F)            // xor 4
            WHT_STAGE(p, 8, 0x201F)            // xor 8
            int pat = lane & 15;
            int row = mt * 16 + r + ((lane >> 4) << 3);  // C/D layout row
            if (pat == 0)      partial[wv][row][0] = p;  // plain sum
            else if (pat == 8) partial[wv][row][1] = p;  // qubit 4
            else if (pat == 4) partial[wv][row][2] = p;  // qubit 5
            else if (pat == 2) partial[wv][row][3] = p;  // qubit 6
            else if (pat == 1) partial[wv][row][4] = p;  // qubit 7
        }
    }
    __syncthreads();

    // --- final reduction over the 16 waves: 64 rows x 8 qubits = 512 outs ---
    {
        int row = tid >> 3, q = tid & 7;
        float acc = 0.0f;
        if (q < 4) {
            int sh = 3 - q;                    // sign bit lives in wv
#pragma unroll
            for (int w2 = 0; w2 < 16; ++w2) {
                float s = partial[w2][row][0];
                acc += ((w2 >> sh) & 1) ? -s : s;
            }
        } else {
            int slot = q - 3;                  // q=4 -> 1, ..., q=7 -> 4
#pragma unroll
            for (int w2 = 0; w2 < 16; ++w2)
                acc += partial[w2][row][slot];
        }
        out[(rb + (tid >> 3)) * NQ + q] = acc;
    }
}

// ---------------------------------------------------------------------------

extern "C" void kernel_launch(void* const* d_in, const int* in_sizes, int n_in,
                              void* d_out, int out_size, void* d_ws, size_t ws_size,
                              hipStream_t stream) {
    const float* x    = (const float*)d_in[0];   // (B, 8)
    const float* conv = (const float*)d_in[1];   // (2, 7, 3)
    const float* pool = (const float*)d_in[2];   // (3, 3)
    float* out        = (float*)d_out;           // (B, 8)
    float* RePlane    = (float*)d_ws;            // 256*256 f32
    float* ImPlane    = RePlane + DIM * DIM;     // 256*256 f32  (512 KB total)

    const int B = in_sizes[0] / NQ;

    build_unitary_kernel<<<DIM, 256, 0, stream>>>(conv, pool, RePlane, ImPlane);
    qcnn_wmma_kernel<<<B / ROWS, 512, 0, stream>>>(x, RePlane, ImPlane, out);
}